// Convolution_81716047774177
// MI455X (gfx1250) — compile-verified
//
#include <hip/hip_runtime.h>
#include <hip/hip_bf16.h>

typedef __attribute__((ext_vector_type(2))) float v2f;
typedef __attribute__((ext_vector_type(8))) float v8f;

#define Hh 48
#define Ww 48
#define HW 2304          // 48*48
#define B_ 8
#define C_ 128
#define KK 9
#define COUT 256
#define KIN 1152         // KK*C_
#define HID 512
#define REGION 12
#define APAD 132         // 16 x 128 A tile, padded row stride (bank-conflict-free)

// ---- workspace layout (floats) ----
#define XT_OFF   0                         // (B, HW, C) transposed x : 2,359,296 floats
#define MEAN_OFF (B_*HW*C_)                // (HW, K, 2)
#define SIG_OFF  (MEAN_OFF + HW*KK*2)      // (HW, K, 2)
#define FL_OFF   (SIG_OFF  + HW*KK*2)      // (HW, K, 2) as int

// ---------------------------------------------------------------------------
// Kernel 0: transpose x (b,c,h,w) -> xt (b, hw, c) so gathers are contiguous
// ---------------------------------------------------------------------------
__global__ __launch_bounds__(256) void xpose_kernel(const float* __restrict__ x,
                                                    float* __restrict__ xt) {
    int idx = blockIdx.x * 256 + threadIdx.x;          // total B*HW*C
    int c = idx & (C_ - 1);
    int s = (idx >> 7) % HW;
    int b = idx / (HW * C_);
    xt[idx] = x[(b * C_ + c) * HW + s];
}

// ---------------------------------------------------------------------------
// Kernel 1: coordinate MLP (batch-independent) -> means / sigmas / floor(means)
// one thread per pixel; all weight loads are wave-uniform (scalar path)
// ---------------------------------------------------------------------------
__global__ __launch_bounds__(256) void mlp_kernel(const float* __restrict__ W1,
                                                  const float* __restrict__ b1,
                                                  const float* __restrict__ W2,
                                                  const float* __restrict__ b2,
                                                  float* __restrict__ ws) {
    int s = blockIdx.x * 256 + threadIdx.x;            // 0..2303
    if (s >= HW) return;
    int y = s / Ww, x = s % Ww;
    float r = (float)y / (float)(Hh - 1);
    float c = (float)x / (float)(Ww - 1);

    float acc[27];
#pragma unroll
    for (int p = 0; p < 27; ++p) acc[p] = b2[p];

    for (int j = 0; j < HID; ++j) {
        float hv = fmaxf(0.0f, r * W1[2 * j] + c * W1[2 * j + 1] + b1[j]);
#pragma unroll
        for (int p = 0; p < 27; ++p) acc[p] += W2[p * HID + j] * hv;
    }

    float scY = r * 0.9999f + 5e-5f;
    float scX = c * 0.9999f + 5e-5f;
    float midY = __logf(scY / (1.0f - scY));
    float midX = __logf(scX / (1.0f - scX));

    float* meanb = ws + MEAN_OFF;
    float* sigb  = ws + SIG_OFF;
    int*   flb   = (int*)(ws + FL_OFF);

#pragma unroll
    for (int k = 0; k < KK; ++k) {
        float zy = midY + 0.1f * acc[2 * k];
        float zx = midX + 0.1f * acc[2 * k + 1];
        float mY = (float)(Hh - 1) / (1.0f + __expf(-zy));
        float mX = (float)(Ww - 1) / (1.0f + __expf(-zx));
        float sp = __logf(1.0f + __expf(acc[18 + k] + 2.0f)) + 0.05f; // softplus+MIN_SIGMA
        float sY = sp * (float)Hh * 0.05f;
        float sX = sp * (float)Ww * 0.05f;
        int o = (s * KK + k) * 2;
        meanb[o] = mY;     meanb[o + 1] = mX;
        sigb[o]  = sY;     sigb[o + 1]  = sX;
        flb[o]   = (int)floorf(mY);
        flb[o+1] = (int)floorf(mX);
    }
}

// ---------------------------------------------------------------------------
// Kernel 2: fused gather + Gaussian-blend + f32 WMMA GEMM + bias + transpose
// grid = B * (HW/16) workgroups of 256 threads (8 waves).
// Each WG: M-tile of 16 positions, full N=256 (2 x 16-wide tiles per wave).
// ---------------------------------------------------------------------------
__global__ __launch_bounds__(256) void fused_kernel(const float* __restrict__ xt,
                                                    const float* __restrict__ Wu,
                                                    const float* __restrict__ bu,
                                                    const int*   __restrict__ gints,
                                                    const int*   __restrict__ roff,
                                                    const float* __restrict__ ws,
                                                    float* __restrict__ out) {
    __shared__ float A_lds[16 * APAD];     // 16 x 128 blended feature tile (padded)
    __shared__ float C_lds[COUT * 17];     // output staging for coalesced store
    __shared__ int   iy_s[128], ix_s[128], lin_s[128];
    __shared__ float prop_s[128], w_s[128];

    const int t    = threadIdx.x;
    const int lane = t & 31;
    const int wv   = t >> 5;               // wave id 0..7
    const int n0   = wv * 32;              // this wave's first N tile

    const int tileIdx = blockIdx.x;
    const int bimg = tileIdx / (HW / 16);
    const int s0   = (tileIdx % (HW / 16)) * 16;

    const float* meanb = ws + MEAN_OFF;
    const float* sigb  = ws + SIG_OFF;
    const int*   flb   = (const int*)(ws + FL_OFF);
    const float* xtb   = xt + (size_t)bimg * HW * C_;

    v8f acc0 = {};
    v8f acc1 = {};

    for (int kk = 0; kk < KK; ++kk) {
        // ---- phase 1a: sample integer coords (threads 0..127 : (p,v)) ----
        int tyR = 0, txR = 0, pR = 0, vR = 0, sR = 0;
        if (t < 128) {
            pR = t >> 3; vR = t & 7; sR = s0 + pR;
            int o  = (sR * KK + kk) * 2;
            int fy = flb[o], fx = flb[o + 1];
            int ty, tx;
            if (vR < 4) {
                ty = fy + (vR >> 1);
                tx = fx + (vR & 1);
            } else if (vR < 6) {
                int base = ((((bimg * HW + sR) * KK + kk) * 2 + (vR - 4)) * 2);
                ty = gints[base]; tx = gints[base + 1];
            } else {
                int base = ((((bimg * HW + sR) * KK + kk) * 2 + (vR - 6)) * 2);
                ty = fy + roff[base]     - REGION / 2;
                tx = fx + roff[base + 1] - REGION / 2;
            }
            ty = ((ty % Hh) + Hh) % Hh;
            tx = ((tx % Ww) + Ww) % Ww;
            tyR = ty; txR = tx;
            iy_s[t] = ty; ix_s[t] = tx;
        }
        __syncthreads();

        // ---- phase 1b: dedup + Gaussian proposal ----
        if (t < 128) {
            bool dup = false;
            for (int v2 = 0; v2 < vR; ++v2)
                dup |= (iy_s[pR * 8 + v2] == tyR) && (ix_s[pR * 8 + v2] == txR);
            int o = (sR * KK + kk) * 2;
            float dy = ((float)tyR - meanb[o])     / sigb[o];
            float dx = ((float)txR - meanb[o + 1]) / sigb[o + 1];
            float pr = dup ? 0.0f : __expf(-0.5f * (dy * dy + dx * dx));
            prop_s[t] = pr;
            lin_s[t]  = tyR * Ww + txR;
        }
        __syncthreads();

        // ---- phase 1c: normalize weights ----
        if (t < 128) {
            float sum = 0.0f;
#pragma unroll
            for (int v2 = 0; v2 < 8; ++v2) sum += prop_s[pR * 8 + v2];
            w_s[t] = prop_s[t] / sum;
        }
        __syncthreads();

        // ---- phase 2: gather + blend 16x128 A-tile (all 256 threads) ----
        {
            int p  = t >> 4;
            int ci = (t & 15) * 8;
            float4 a0 = {0, 0, 0, 0}, a1 = {0, 0, 0, 0};
#pragma unroll
            for (int v = 0; v < 8; ++v) {
                float wt = w_s[p * 8 + v];
                const float4* src =
                    (const float4*)(xtb + (size_t)lin_s[p * 8 + v] * C_ + ci);
                float4 u0 = src[0], u1 = src[1];
                a0.x += wt * u0.x; a0.y += wt * u0.y; a0.z += wt * u0.z; a0.w += wt * u0.w;
                a1.x += wt * u1.x; a1.y += wt * u1.y; a1.z += wt * u1.z; a1.w += wt * u1.w;
            }
            float* dst = A_lds + p * APAD + ci;
            dst[0] = a0.x; dst[1] = a0.y; dst[2] = a0.z; dst[3] = a0.w;
            dst[4] = a1.x; dst[5] = a1.y; dst[6] = a1.z; dst[7] = a1.w;
        }
        __syncthreads();

        // ---- phase 3: WMMA f32 16x16x4 over the 128-deep chunk ----
        {
            const int m    = lane & 15;
            const int koff = (lane >> 4) * 2;       // lanes 16-31 carry K+2,K+3
            const float* arow  = A_lds + m * APAD + koff;
            const float* brow0 = Wu + (size_t)(n0 + (lane & 15)) * KIN + kk * C_ + koff;
            const float* brow1 = brow0 + (size_t)16 * KIN;
#pragma unroll
            for (int k0 = 0; k0 < C_; k0 += 4) {
                v2f a  = *(const v2f*)(arow + k0);
                v2f b0 = *(const v2f*)(brow0 + k0);
                v2f b1 = *(const v2f*)(brow1 + k0);
                acc0 = __builtin_amdgcn_wmma_f32_16x16x4_f32(
                    false, a, false, b0, (short)0, acc0, false, false);
                acc1 = __builtin_amdgcn_wmma_f32_16x16x4_f32(
                    false, a, false, b1, (short)0, acc1, false, false);
            }
        }
        __syncthreads();   // protect A_lds / weight buffers before next head
    }

    // ---- epilogue: bias, stage through LDS, coalesced (b,cout,h,w) store ----
    {
        int nA = n0 + (lane & 15);
        int nB = nA + 16;
        float buA = bu[nA], buB = bu[nB];
        int mbase = (lane >> 4) * 8;
#pragma unroll
        for (int r = 0; r < 8; ++r) {
            C_lds[nA * 17 + mbase + r] = acc0[r] + buA;
            C_lds[nB * 17 + mbase + r] = acc1[r] + buB;
        }
    }
    __syncthreads();
    {
        int n = t;  // 0..255
        float* dst = out + ((size_t)(bimg * COUT + n) * HW + s0);
#pragma unroll
        for (int i = 0; i < 4; ++i) {
            float4 v;
            v.x = C_lds[n * 17 + 4 * i + 0];
            v.y = C_lds[n * 17 + 4 * i + 1];
            v.z = C_lds[n * 17 + 4 * i + 2];
            v.w = C_lds[n * 17 + 4 * i + 3];
            ((float4*)dst)[i] = v;
        }
    }
}

// ---------------------------------------------------------------------------
extern "C" void kernel_launch(void* const* d_in, const int* in_sizes, int n_in,
                              void* d_out, int out_size, void* d_ws, size_t ws_size,
                              hipStream_t stream) {
    const float* x  = (const float*)d_in[0];
    const float* W1 = (const float*)d_in[1];
    const float* b1 = (const float*)d_in[2];
    const float* W2 = (const float*)d_in[3];
    const float* b2 = (const float*)d_in[4];
    const float* Wu = (const float*)d_in[5];
    const float* bu = (const float*)d_in[6];
    const int* gints = (const int*)d_in[7];
    const int* roff  = (const int*)d_in[8];
    float* out = (float*)d_out;
    float* ws  = (float*)d_ws;

    // 0) transpose x -> (b, hw, c)
    int nT = B_ * HW * C_;
    xpose_kernel<<<nT / 256, 256, 0, stream>>>(x, ws + XT_OFF);

    // 1) coordinate MLP (batch-independent)
    mlp_kernel<<<(HW + 255) / 256, 256, 0, stream>>>(W1, b1, W2, b2, ws);

    // 2) fused gather/blend/GEMM
    int nWG = B_ * (HW / 16);  // 1152 workgroups
    fused_kernel<<<nWG, 256, 0, stream>>>(ws + XT_OFF, Wu, bu, gints, roff, ws, out);
}